// HybridModel_22548578304629
// MI455X (gfx1250) — compile-verified
//
#include <hip/hip_runtime.h>
#include <hip/hip_bf16.h>

typedef __attribute__((ext_vector_type(2))) float v2f;
typedef __attribute__((ext_vector_type(8))) float v8f;

#ifndef HDIM
#define HDIM 64
#endif

// ---------------------------------------------------------------------------
// Utility fill
// ---------------------------------------------------------------------------
__global__ void fill_kernel(float* __restrict__ p, float v, long n) {
    long i = (long)blockIdx.x * blockDim.x + threadIdx.x;
    if (i < n) p[i] = v;
}

// deg[dst] += 1 per edge (self-loop handled by init fill of 1.0)
__global__ void deg_edge_kernel(const int* __restrict__ ei, float* __restrict__ deg, int E) {
    int e = blockIdx.x * blockDim.x + threadIdx.x;
    if (e < E) unsafeAtomicAdd(&deg[ei[E + e]], 1.0f);
}

__global__ void rsqrt_kernel(const float* __restrict__ deg, float* __restrict__ dis, int n) {
    int i = blockIdx.x * blockDim.x + threadIdx.x;
    if (i < n) dis[i] = rsqrtf(deg[i]);   // deg >= 1 always (self-loop)
}

// ---------------------------------------------------------------------------
// GCN layer 1: h1 = relu( scatter(x[src]*norm * Wg1) + bg1 )
// ---------------------------------------------------------------------------
// self-loop contribution: out[n,c] = x[n]*dis[n]^2 * Wg1[c]
__global__ void l1_init_kernel(const float* __restrict__ x, const float* __restrict__ dis,
                               const float* __restrict__ Wg1, float* __restrict__ out, long NH) {
    long i = (long)blockIdx.x * blockDim.x + threadIdx.x;
    if (i >= NH) return;
    long n = i >> 6; int c = (int)(i & 63);
    float d = dis[n];
    out[i] = x[n] * d * d * Wg1[c];
}

// one wave (32 lanes) per edge; lane covers channels {2*lane, 2*lane+1}
__global__ __launch_bounds__(256) void l1_edge_kernel(
    const int* __restrict__ ei, const float* __restrict__ x, const float* __restrict__ dis,
    const float* __restrict__ Wg1, float* __restrict__ out, int E) {
    int wave = blockIdx.x * (blockDim.x >> 5) + (threadIdx.x >> 5);
    int lane = threadIdx.x & 31;
    if (wave >= E) return;
    int s = ei[wave];
    int d = ei[E + wave];
    float xw = x[s] * dis[s] * dis[d];
    float2 wv = ((const float2*)Wg1)[lane];           // global_load_b64
    float* od = out + (long)d * HDIM + 2 * lane;
    unsafeAtomicAdd(&od[0], xw * wv.x);               // global_atomic_add_f32
    unsafeAtomicAdd(&od[1], xw * wv.y);
}

__global__ void bias_relu_kernel(float* __restrict__ h, const float* __restrict__ b, long NH) {
    long i = (long)blockIdx.x * blockDim.x + threadIdx.x;
    if (i >= NH) return;
    h[i] = fmaxf(h[i] + b[i & 63], 0.0f);
}

// ---------------------------------------------------------------------------
// GCN layer 2 propagation (hw already = h1 @ Wg2)
// ---------------------------------------------------------------------------
__global__ void l2_init_kernel(const float* __restrict__ hw, const float* __restrict__ dis,
                               float* __restrict__ out, long NH) {
    long i = (long)blockIdx.x * blockDim.x + threadIdx.x;
    if (i >= NH) return;
    long n = i >> 6;
    float d = dis[n];
    out[i] = hw[i] * d * d;
}

__global__ __launch_bounds__(256) void l2_edge_kernel(
    const int* __restrict__ ei, const float* __restrict__ dis,
    const float* __restrict__ hw, float* __restrict__ out, int E) {
    int wave = blockIdx.x * (blockDim.x >> 5) + (threadIdx.x >> 5);
    int lane = threadIdx.x & 31;
    if (wave >= E) return;
    int s = ei[wave];
    int d = ei[E + wave];
    float w = dis[s] * dis[d];
    float2 hv = ((const float2*)(hw + (long)s * HDIM))[lane];  // coalesced b64 gather
    float* od = out + (long)d * HDIM + 2 * lane;
    unsafeAtomicAdd(&od[0], hv.x * w);
    unsafeAtomicAdd(&od[1], hv.y * w);
}

// ---------------------------------------------------------------------------
// Mean pool
// ---------------------------------------------------------------------------
__global__ void pool_sum_kernel(const float* __restrict__ h, const int* __restrict__ batch,
                                float* __restrict__ sums, long NH) {
    long i = (long)blockIdx.x * blockDim.x + threadIdx.x;
    if (i >= NH) return;
    long n = i >> 6; int c = (int)(i & 63);
    unsafeAtomicAdd(&sums[(long)batch[n] * HDIM + c], h[i]);
}

__global__ void pool_cnt_kernel(const int* __restrict__ batch, float* __restrict__ cnt, int N) {
    int i = blockIdx.x * blockDim.x + threadIdx.x;
    if (i < N) unsafeAtomicAdd(&cnt[batch[i]], 1.0f);
}

// graph_out -> combined[:, H:2H]
__global__ void graph_out_kernel(const float* __restrict__ sums, const float* __restrict__ cnt,
                                 float* __restrict__ combined, int BH) {
    int i = blockIdx.x * blockDim.x + threadIdx.x;
    if (i >= BH) return;
    int b = i >> 6; int c = i & 63;
    combined[(long)b * (2 * HDIM) + HDIM + c] = sums[i] / fmaxf(cnt[b], 1.0f);
}

// ---------------------------------------------------------------------------
// WMMA f32 GEMM:  C[M,Ncols](ldc) = act(A[M,K] @ W[K,Ncols] + bias)
// One wave computes a 16 x (16*NT) tile with V_WMMA_F32_16X16X4_F32.
// A frag reused across NT column-tiles each k-step.
// All memory indexing is uniform-base + unsigned 32-bit per-lane offsets so
// the backend emits saddr-form global_load (GVS) instead of flat_load with
// per-lane 64-bit pointer math.
// OOB columns handled by CLAMPING the B read offset (a bad B column only
// corrupts that same unstored D column); stores are guarded.
// A frag (16x4, 2 VGPR/lane): v0 = A[M=l%16, K=k+2*(l/16)], v1 = +1   (ISA 7.12.2)
// B frag (4x16): v0 = W[K=k+2*(l/16), N=col], v1 = +1
// C/D (16x16, 8 VGPR): vj = C[M=j+8*(l/16), N=col]
// ---------------------------------------------------------------------------
template <int NT>
__global__ __launch_bounds__(32) void wmma_gemm_f32(
    const float* __restrict__ A, const float* __restrict__ W,
    const float* __restrict__ bias, float* __restrict__ C,
    unsigned M, unsigned K, unsigned Ncols, unsigned ldc, int do_relu) {
    const unsigned lane = threadIdx.x;
    const unsigned half = lane >> 4;       // 0: lanes 0-15, 1: lanes 16-31
    const unsigned l15  = lane & 15;
    const unsigned row0 = blockIdx.x * 16;
    const unsigned n0   = blockIdx.y * (16 * NT);

    v8f acc[NT] = {};

    unsigned aoff = (row0 + l15) * K + 2 * half;      // element offset into A

    unsigned cols[NT];
    unsigned woff[NT];
#pragma unroll
    for (int t = 0; t < NT; ++t) {
        unsigned c = n0 + t * 16 + l15;
        cols[t] = c;
        unsigned cc = (c < Ncols) ? c : (Ncols - 1);  // clamp: garbage only hits unstored col
        woff[t] = 2 * half * Ncols + cc;
    }
    const unsigned wstep = 4 * Ncols;

    for (unsigned k = 0; k < K; k += 4) {
        v2f a;
        a[0] = A[aoff];               // merges into global_load_b64 (saddr form)
        a[1] = A[aoff + 1];
        aoff += 4;
#pragma unroll
        for (int t = 0; t < NT; ++t) {
            v2f b;
            b[0] = W[woff[t]];
            b[1] = W[woff[t] + Ncols];
            woff[t] += wstep;
            acc[t] = __builtin_amdgcn_wmma_f32_16x16x4_f32(
                /*neg_a=*/false, a, /*neg_b=*/false, b,
                /*c_mod=*/(short)0, acc[t], /*reuse_a=*/false, /*reuse_b=*/false);
        }
    }

#pragma unroll
    for (int t = 0; t < NT; ++t) {
        unsigned col = cols[t];
        if (col < Ncols) {
            float bv = bias ? bias[col] : 0.0f;
#pragma unroll
            for (unsigned j = 0; j < 8; ++j) {
                unsigned r = row0 + j + 8 * half;     // M always multiple of 16 here
                float v = acc[t][j] + bv;
                if (do_relu) v = fmaxf(v, 0.0f);
                C[r * ldc + col] = v;
            }
        }
    }
}

// ---------------------------------------------------------------------------
// Host launcher
// ---------------------------------------------------------------------------
extern "C" void kernel_launch(void* const* d_in, const int* in_sizes, int n_in,
                              void* d_out, int out_size, void* d_ws, size_t ws_size,
                              hipStream_t stream) {
    const float* tabular = (const float*)d_in[0];
    const float* x       = (const float*)d_in[1];
    const int*   ei      = (const int*)d_in[2];
    const int*   batch   = (const int*)d_in[3];
    const float* Wt1 = (const float*)d_in[4];  const float* bt1 = (const float*)d_in[5];
    const float* Wt2 = (const float*)d_in[6];  const float* bt2 = (const float*)d_in[7];
    const float* Wg1 = (const float*)d_in[8];  const float* bg1 = (const float*)d_in[9];
    const float* Wg2 = (const float*)d_in[10]; const float* bg2 = (const float*)d_in[11];
    const float* Wf1 = (const float*)d_in[12]; const float* bf1 = (const float*)d_in[13];
    const float* Wf2 = (const float*)d_in[14]; const float* bf2 = (const float*)d_in[15];
    float* out = (float*)d_out;

    const int N  = in_sizes[1];           // 200000
    const int E  = in_sizes[2] / 2;       // 6400000
    const int H  = in_sizes[5];           // 64
    const int TD = in_sizes[4] / H;       // 128
    const int B  = in_sizes[0] / TD;      // 1024
    const int OC = in_sizes[15];          // 2
    const long NH = (long)N * H;

    // workspace layout (floats)
    float* ws = (float*)d_ws;
    float* deg      = ws;  ws += N;
    float* dis      = ws;  ws += N;
    float* bufA     = ws;  ws += NH;           // h1 -> out2 accumulator -> h2
    float* bufB     = ws;  ws += NH;           // hw2 = h1 @ Wg2
    float* tabmid   = ws;  ws += (long)B * H;
    float* combined = ws;  ws += (long)B * 2 * H;
    float* sums     = ws;  ws += (long)B * H;
    float* cnt      = ws;  ws += B;
    float* fmid     = ws;  ws += (long)B * H;

    const int T = 256;
    dim3 blk(T);
    auto nb = [](long n, int t) { return (unsigned)((n + t - 1) / t); };

    // 1) degrees (self-loop => init 1.0) and symmetric-norm factor dis = rsqrt(deg)
    fill_kernel<<<nb(N, T), blk, 0, stream>>>(deg, 1.0f, N);
    deg_edge_kernel<<<nb(E, T), blk, 0, stream>>>(ei, deg, E);
    rsqrt_kernel<<<nb(N, T), blk, 0, stream>>>(deg, dis, N);

    // 2) GCN layer 1
    l1_init_kernel<<<nb(NH, T), blk, 0, stream>>>(x, dis, Wg1, bufA, NH);
    l1_edge_kernel<<<nb((long)E * 32, T), blk, 0, stream>>>(ei, x, dis, Wg1, bufA, E);
    bias_relu_kernel<<<nb(NH, T), blk, 0, stream>>>(bufA, bg1, NH);

    // 3) hw2 = h1 @ Wg2  (WMMA, N x 64 x 64; one wave per 16x64 row-panel)
    {
        dim3 g(N / 16, 1);
        wmma_gemm_f32<4><<<g, dim3(32), 0, stream>>>(bufA, Wg2, nullptr, bufB, N, H, H, H, 0);
    }

    // 4) GCN layer 2 propagation
    l2_init_kernel<<<nb(NH, T), blk, 0, stream>>>(bufB, dis, bufA, NH);
    l2_edge_kernel<<<nb((long)E * 32, T), blk, 0, stream>>>(ei, dis, bufB, bufA, E);
    bias_relu_kernel<<<nb(NH, T), blk, 0, stream>>>(bufA, bg2, NH);

    // 5) mean pool over graphs
    fill_kernel<<<nb((long)B * H, T), blk, 0, stream>>>(sums, 0.0f, (long)B * H);
    fill_kernel<<<nb(B, T), blk, 0, stream>>>(cnt, 0.0f, B);
    pool_sum_kernel<<<nb(NH, T), blk, 0, stream>>>(bufA, batch, sums, NH);
    pool_cnt_kernel<<<nb(N, T), blk, 0, stream>>>(batch, cnt, N);
    graph_out_kernel<<<nb((long)B * H, T), blk, 0, stream>>>(sums, cnt, combined, B * H);

    // 6) tabular MLP: tab = relu(tabular@Wt1+bt1)@Wt2+bt2 -> combined[:, :H]
    {
        dim3 g1(B / 16, 1);
        wmma_gemm_f32<4><<<g1, dim3(32), 0, stream>>>(tabular, Wt1, bt1, tabmid, B, TD, H, H, 1);
        wmma_gemm_f32<4><<<g1, dim3(32), 0, stream>>>(tabmid, Wt2, bt2, combined, B, H, H, 2 * H, 0);
    }

    // 7) final MLP: out = relu(combined@Wf1+bf1)@Wf2+bf2
    {
        dim3 g1(B / 16, 1);
        wmma_gemm_f32<4><<<g1, dim3(32), 0, stream>>>(combined, Wf1, bf1, fmid, B, 2 * H, H, H, 1);
        dim3 g2(B / 16, 1);
        wmma_gemm_f32<1><<<g2, dim3(32), 0, stream>>>(fmid, Wf2, bf2, out, B, H, OC, OC, 0);
    }
}